// KDE_66812511257397
// MI455X (gfx1250) — compile-verified
//
#include <hip/hip_runtime.h>
#include <hip/hip_bf16.h>

typedef __attribute__((ext_vector_type(2))) float v2f;
typedef __attribute__((ext_vector_type(8))) float v8f;

#define NDIM_GRID 20
#define NY_NZ     400     // 20*20
#define NPTS      8000
#define NTILES    500     // 8000 / 16
#define LOG2E     1.4426950408889634f

__device__ __forceinline__ void decode_coord(int i, float& x, float& y, float& z) {
    int ix = i / NY_NZ;
    int rem = i - ix * NY_NZ;
    int iy = rem / NDIM_GRID;
    int iz = rem - iy * NDIM_GRID;
    x = (float)ix; y = (float)iy; z = (float)iz;
}

// One wave per 16-row block; loops over all 500 column tiles.
// The WMMA produces the *complete* base-2 exponent in one op:
//   D = A*B + C = log2e*( x_a^T Ainv x_b - 0.5*q_a - 0.5*q_b )
//   A row m = ( L*g0, L*g1, L*g2, -0.5*L )   (L = log2e, g = Ainv x_a)
//   B col n = ( x,    y,    z,    q_b    )   <- ds_load_b64 straight into regs
//   C[m,*]  = -0.5*L*q_a[m]                  (row bias, loop-invariant)
// Tail per element: raw v_exp_f32 -> fmac. No clamp needed: off-diagonal
// d2 >= ~0.5 analytically; diagonal rounding error (~1e-4 in the exponent)
// perturbs one term of a large sum by <1e-4 relative -> negligible.
__global__ void __launch_bounds__(64) kde_main(
        const float* __restrict__ p,     // space_probs, 8000
        const float* __restrict__ Ainv,  // cov_inv, 9 floats row-major
        float* __restrict__ kde)         // 8000 (unnormalized out)
{
    __shared__ float4 bq[NPTS];          // (x, y, z, q) per column: 128 KB
    __shared__ float  pp[NPTS];          // p per column:            32 KB

    const int tid  = threadIdx.x;
    const int wave = (blockIdx.x * blockDim.x + tid) >> 5;
    const int lane = tid & 31;
    const int half = lane >> 4;          // 0: lanes 0-15, 1: lanes 16-31
    const int l15  = lane & 15;

    // 3x3 inverse covariance (uniform across wave)
    const float a00 = Ainv[0], a01 = Ainv[1], a02 = Ainv[2];
    const float a10 = Ainv[3], a11 = Ainv[4], a12 = Ainv[5];
    const float a20 = Ainv[6], a21 = Ainv[7], a22 = Ainv[8];

    // ---- Stage per-column (x,y,z,q) and p into LDS (once per workgroup).
    for (int i = tid; i < NPTS; i += 64) {
        float cx, cy, cz; decode_coord(i, cx, cy, cz);
        float b0 = a00 * cx + a01 * cy + a02 * cz;
        float b1 = a10 * cx + a11 * cy + a12 * cz;
        float b2 = a20 * cx + a21 * cy + a22 * cz;
        float q  = cx * b0 + cy * b1 + cz * b2;
        bq[i] = make_float4(cx, cy, cz, q);
        pp[i] = p[i];
    }
    __syncthreads();

    // ---- Build WMMA A operand for this wave's 16 rows.
    // f32 16x4 A layout: lane half0 holds K=0,1 ; half1 holds K=2,3.
    const int row = wave * 16 + l15;
    float rx, ry, rz; decode_coord(row, rx, ry, rz);
    const float g0 = a00 * rx + a01 * ry + a02 * rz;
    const float g1 = a10 * rx + a11 * ry + a12 * rz;
    const float g2 = a20 * rx + a21 * ry + a22 * rz;
    const float qrow = rx * g0 + ry * g1 + rz * g2;   // q_a (unscaled)
    v2f amat;
    amat.x = half ? (LOG2E * g2) : (LOG2E * g0);
    amat.y = half ? (-0.5f * LOG2E) : (LOG2E * g1);

    // C operand: -0.5*log2e*q_a for rows M = r + 8*half (C/D layout: value
    // depends only on the row, so it is uniform across the 16 N-lanes).
    v8f cvec;
#pragma unroll
    for (int r = 0; r < 8; ++r)
        cvec[r] = (-0.5f * LOG2E) * __shfl(qrow, (lane & 16) + half * 8 + r, 32);

    float acc[8] = {};

    // Per-lane LDS view: column col = t*16 + l15; this lane's B half is the
    // float2 at index col*2 + half of bq -> (x,y) for half0, (z,q) for half1.
    const v2f*   bbase = (const v2f*)bq + (l15 * 2 + half);
    const float* pbase = pp + l15;

#pragma unroll 2
    for (int t = 0; t < NTILES; ++t) {
        const v2f   bmat = bbase[t * 32];   // B operand pair, no VALU needed
        const float pb   = pbase[t * 16];

        v8f c = __builtin_amdgcn_wmma_f32_16x16x4_f32(
                false, amat, false, bmat, (short)0, cvec, false, false);

#pragma unroll
        for (int r = 0; r < 8; ++r)
            acc[r] = fmaf(__builtin_amdgcn_exp2f(c[r]), pb, acc[r]);
    }

    // Row sums: reduce across the 16 lanes of each half, store 16 rows.
#pragma unroll
    for (int r = 0; r < 8; ++r) {
        float v = acc[r];
        v += __shfl_xor(v, 1, 32);
        v += __shfl_xor(v, 2, 32);
        v += __shfl_xor(v, 4, 32);
        v += __shfl_xor(v, 8, 32);
        if (l15 == 0)
            kde[wave * 16 + r + 8 * half] = v;   // M = r + 8*half
    }
}

// Single-block normalize: out /= sum(out). (1/sqrt(2pi) coefficient cancels.)
__global__ void kde_norm(float* __restrict__ kde) {
    __shared__ float red[256];
    float s = 0.0f;
    for (int i = threadIdx.x; i < NPTS; i += 256) s += kde[i];
    red[threadIdx.x] = s;
    __syncthreads();
    for (int off = 128; off > 0; off >>= 1) {
        if (threadIdx.x < off) red[threadIdx.x] += red[threadIdx.x + off];
        __syncthreads();
    }
    const float inv = 1.0f / red[0];
    for (int i = threadIdx.x; i < NPTS; i += 256) kde[i] *= inv;
}

extern "C" void kernel_launch(void* const* d_in, const int* in_sizes, int n_in,
                              void* d_out, int out_size, void* d_ws, size_t ws_size,
                              hipStream_t stream) {
    const float* p    = (const float*)d_in[0];   // space_probs (8000)
    const float* ainv = (const float*)d_in[1];   // cov_inv (9)
    float* out = (float*)d_out;                  // 8000 f32

    // 500 waves (one per 16-row block), 2 waves per 64-thread block -> 250 blocks.
    kde_main<<<250, 64, 0, stream>>>(p, ainv, out);
    kde_norm<<<1, 256, 0, stream>>>(out);
}